// SIMcLoss_62929860821405
// MI455X (gfx1250) — compile-verified
//
#include <hip/hip_runtime.h>

// ---------------------------------------------------------------------------
// simloss = (||Xn Xn^T||_F^2 - N) / (N^2 - N)  computed as ||Xn^T Xn||_F^2
// Xn = X / max(||row||, eps),  X: [8192, 256] fp32
// ---------------------------------------------------------------------------

typedef float v2f __attribute__((ext_vector_type(2)));
typedef float v8f __attribute__((ext_vector_type(8)));

#define NROWS 8192
#define DDIM  256
#define KSPLIT 4           // blockIdx.z K-partitions
#define WAVES 8            // waves per block (256 threads, wave32)
#define EPS_F 1e-8f

// ---- Kernel 1: per-row inverse norms (one wave per row) --------------------
__global__ __launch_bounds__(256)
void SIMc_rownorm(const float* __restrict__ x, float* __restrict__ invn) {
    const int lane = threadIdx.x & 31;
    const int wave = threadIdx.x >> 5;
    const int row  = blockIdx.x * WAVES + wave;           // 1024 blocks * 8 waves
    const float* r = x + (size_t)row * DDIM;
    float s = 0.f;
#pragma unroll
    for (int i = 0; i < DDIM; i += 32) {
        float v = r[i + lane];
        s = fmaf(v, v, s);
    }
#pragma unroll
    for (int off = 16; off; off >>= 1) s += __shfl_xor(s, off, 32);
    if (lane == 0) invn[row] = 1.0f / fmaxf(sqrtf(s), EPS_F);
}

// ---- Kernel 2: C_partial[z] += Xn^T Xn over one K slice (fp32 WMMA) --------
// One block -> one 16x16 output tile (tm,tn) for K-slice z.
// Wave w covers K in [z*2048 + w*256, +256) with 64 v_wmma_f32_16x16x4_f32.
__global__ __launch_bounds__(256)
void SIMc_gram(const float* __restrict__ x, const float* __restrict__ invn,
               float* __restrict__ Cpart) {
    __shared__ float part[WAVES * 256];                   // 8 KB LDS

    const int tid   = threadIdx.x;
    const int lane  = tid & 31;
    const int wave  = tid >> 5;
    const int tm    = blockIdx.x;                         // 0..15
    const int tn    = blockIdx.y;                         // 0..15
    const int mcol  = tm * 16 + (lane & 15);              // A: M = lane%16
    const int ncol  = tn * 16 + (lane & 15);              // B: N = lane%16
    const int khalf = (lane >> 4) << 1;                   // K offset 0 or 2
    const int kbase = blockIdx.z * (NROWS / KSPLIT) + wave * 256;

    v8f acc = {0.f, 0.f, 0.f, 0.f, 0.f, 0.f, 0.f, 0.f};

    for (int kk = 0; kk < 256; kk += 4) {
        const int k0 = kbase + kk + khalf;
        const float s0 = invn[k0];
        const float s1 = invn[k0 + 1];
        const float* p0 = x + (size_t)k0 * DDIM;
        const float* p1 = p0 + DDIM;
        v2f a, b;
        a.x = p0[mcol] * s0;  a.y = p1[mcol] * s1;        // A[m][k] = Xn[k][m]
        b.x = p0[ncol] * s0;  b.y = p1[ncol] * s1;        // B[k][n] = Xn[k][n]
        // 8 args: (neg_a, A, neg_b, B, c_mod, C, reuse_a, reuse_b)
        acc = __builtin_amdgcn_wmma_f32_16x16x4_f32(
            false, a, false, b, (short)0, acc, false, false);
    }

    // C/D layout: VGPR r, lanes 0-15 -> M=r ; lanes 16-31 -> M=r+8 ; N=lane%16
    const int col   = lane & 15;
    const int rbase = (lane >> 4) << 3;
#pragma unroll
    for (int r = 0; r < 8; ++r)
        part[wave * 256 + (rbase + r) * 16 + col] = acc[r];
    __syncthreads();

    // Fixed-order cross-wave reduction (deterministic), then one store.
    float sum = 0.f;
#pragma unroll
    for (int w = 0; w < WAVES; ++w) sum += part[w * 256 + tid];

    const int rloc = tid >> 4, cloc = tid & 15;
    Cpart[((size_t)blockIdx.z << 16) +
          (size_t)(tm * 16 + rloc) * DDIM + (tn * 16 + cloc)] = sum;
}

// ---- Kernel 3: simloss = (sum(C^2) - N) / (N^2 - N) ------------------------
__global__ __launch_bounds__(256)
void SIMc_reduce(const float* __restrict__ Cpart, float* __restrict__ out) {
    __shared__ float red[WAVES];
    const int tid = threadIdx.x;
    float s = 0.f;
    for (int i = tid; i < DDIM * DDIM; i += 256) {
        float v = Cpart[i] + Cpart[i + (1 << 16)] +
                  Cpart[i + (2 << 16)] + Cpart[i + (3 << 16)];
        s = fmaf(v, v, s);
    }
#pragma unroll
    for (int off = 16; off; off >>= 1) s += __shfl_xor(s, off, 32);
    if ((tid & 31) == 0) red[tid >> 5] = s;
    __syncthreads();
    if (tid == 0) {
        float t = 0.f;
#pragma unroll
        for (int w = 0; w < WAVES; ++w) t += red[w];
        const float nf = (float)NROWS;
        out[0] = (t - nf) / (nf * nf - nf);
    }
}

// ---------------------------------------------------------------------------
extern "C" void kernel_launch(void* const* d_in, const int* in_sizes, int n_in,
                              void* d_out, int out_size, void* d_ws, size_t ws_size,
                              hipStream_t stream) {
    (void)in_sizes; (void)n_in; (void)out_size; (void)ws_size;
    const float* x = (const float*)d_in[0];
    float* invn  = (float*)d_ws;                 // 8192 floats   (32 KB)
    float* Cpart = invn + NROWS;                 // 4 * 65536 floats (1 MB)
    float* out   = (float*)d_out;

    SIMc_rownorm<<<NROWS / WAVES, 256, 0, stream>>>(x, invn);
    SIMc_gram<<<dim3(16, 16, KSPLIT), 256, 0, stream>>>(x, invn, Cpart);
    SIMc_reduce<<<1, 256, 0, stream>>>(Cpart, out);
}